// Router_45526653337602
// MI455X (gfx1250) — compile-verified
//
#include <hip/hip_runtime.h>
#include <hip/hip_bf16.h>
#include <math.h>

// ---- problem constants (match reference) ----
#define BB   2
#define SS   2048
#define DD   256
#define NQK  8192
#define NVN  8192
#define NCPS 32
#define MPC  16
#define NC   144            // 9*MPC candidates
#define TOK  (BB*SS)        // 4096 tokens
#define GSZ  ((size_t)TOK*NC) // 589824 per gate

typedef __attribute__((ext_vector_type(2))) float v2f;
typedef __attribute__((ext_vector_type(8))) float v8f;

__device__ __forceinline__ float gate_val(float s, float tau) {
    float raw  = s - tau;
    float gate = raw > 0.0f ? raw : 1e-8f * expf(raw);
    return expf(gate) - 1.0f;           // mirrors exp(gate)-1.0
}

__global__ __launch_bounds__(256) void router_fwd(
    const float* __restrict__ x,
    const float* __restrict__ qkN,
    const float* __restrict__ vN,
    const float* __restrict__ npos,      // (NQK+NVN, 2)
    const float* __restrict__ Wpq, const float* __restrict__ bpq,
    const float* __restrict__ Wpv, const float* __restrict__ bpv,
    const float* __restrict__ Wtau, const float* __restrict__ btau,
    const int*   __restrict__ cmQ,
    const int*   __restrict__ cmV,
    float* __restrict__ out,             // 3*GSZ gates (+loss slot written by kernel 2)
    float* __restrict__ ws)              // 4*TOK partials
{
    __shared__ __align__(16) float xs[DD];
    __shared__ float proj[7];            // qk_pos(2), v_pos(2), tau(3)
    __shared__ int   ciQ[NC], ciV[NC];
    __shared__ float sQ[NC], sV[NC];
    __shared__ float egQ[NC], egK[NC], egV[NC];
    __shared__ float thrQ, thrV;
    __shared__ float scl[3];
    __shared__ float4 red[256];

    const int t     = threadIdx.x;
    const int lane  = t & 31;
    const int wave  = t >> 5;
    const int token = blockIdx.x;

    // ---- phase 1: stage x, compute the 7 projections (wave-shuffle tree, deterministic)
    xs[t] = x[(size_t)token * DD + t];
    __syncthreads();

    if (wave < 7) {
        const float* W; int stride, col; float bias;
        if (wave < 2)      { W = Wpq;  stride = 2; col = wave;     bias = bpq[col]; }
        else if (wave < 4) { W = Wpv;  stride = 2; col = wave - 2; bias = bpv[col]; }
        else               { W = Wtau; stride = 3; col = wave - 4; bias = btau[col]; }
        float acc = 0.0f;
        for (int d = lane; d < DD; d += 32) acc += xs[d] * W[d * stride + col];
        #pragma unroll
        for (int off = 16; off > 0; off >>= 1) acc += __shfl_xor(acc, off, 32);
        if (lane == 0) proj[wave] = acc + bias;
    }
    __syncthreads();

    // ---- phase 2: spatial hash candidate lookup (9 neighbor cells x 16 slots)
    if (t < NC) {
        const int offi = t >> 4, slot = t & 15;
        const int ox = offi / 3 - 1, oy = offi % 3 - 1;
        {
            float fx = (proj[0] + 4.0f) * (1.0f / 8.0f) * NCPS;
            float fy = (proj[1] + 4.0f) * (1.0f / 8.0f) * NCPS;
            int cx = (int)fminf(fmaxf(fx, 0.0f), 31.0f);
            int cy = (int)fminf(fmaxf(fy, 0.0f), 31.0f);
            int nx = min(max(cx + ox, 0), NCPS - 1);
            int ny = min(max(cy + oy, 0), NCPS - 1);
            ciQ[t] = cmQ[(nx * NCPS + ny) * MPC + slot];
        }
        {
            float fx = (proj[2] + 4.0f) * (1.0f / 8.0f) * NCPS;
            float fy = (proj[3] + 4.0f) * (1.0f / 8.0f) * NCPS;
            int cx = (int)fminf(fmaxf(fx, 0.0f), 31.0f);
            int cy = (int)fminf(fmaxf(fy, 0.0f), 31.0f);
            int nx = min(max(cx + ox, 0), NCPS - 1);
            int ny = min(max(cy + oy, 0), NCPS - 1);
            ciV[t] = cmV[(nx * NCPS + ny) * MPC + slot];
        }
    }
    __syncthreads();

    // ---- phase 3: candidate scores via V_WMMA_F32_16X16X4_F32
    // One wave per 16-candidate tile; B columns all replicate x, so D[:,n] are
    // identical score vectors. 18 tasks = {qk,v} x 9 tiles.
    for (int task = wave; task < 18; task += 8) {
        const int side = task >= 9;
        const int tile = side ? task - 9 : task;
        const float* nb = side ? vN  : qkN;
        const int*   ci = side ? ciV : ciQ;
        float*       sA = side ? sV  : sQ;

        const int m    = lane & 15;
        const int half = lane >> 4;
        int row = ci[tile * 16 + m];
        if (row < 0) row = 0;                        // invalids masked later
        const float* ap = nb + (size_t)row * DD + 2 * half;
        const float* bp = xs + 2 * half;

        v8f c = {};
        #pragma unroll 8
        for (int k0 = 0; k0 < DD; k0 += 4) {
            v2f a = *(const v2f*)(ap + k0);          // A: lanes0-15 K0,K1 / 16-31 K2,K3
            v2f b = *(const v2f*)(bp + k0);          // B: same K split, all N columns = x
            c = __builtin_amdgcn_wmma_f32_16x16x4_f32(
                    false, a, false, b, (short)0, c, false, false);
        }
        if (m == 0) {                                 // lane 0 -> M 0..7, lane 16 -> M 8..15
            const int b0 = tile * 16 + 8 * half;
            #pragma unroll
            for (int r = 0; r < 8; ++r) sA[b0 + r] = c[r];
        }
    }
    __syncthreads();

    // ---- phase 4: mask invalid candidates
    if (t < NC) {
        if (ciQ[t] < 0) sQ[t] = -1e9f;
        if (ciV[t] < 0) sV[t] = -1e9f;
    }
    __syncthreads();

    // ---- phase 5: k-th largest thresholds via stable ranking
    // (gate transform is monotone in score, tau is uniform -> threshold on s)
    if (t < NC) {
        float st = sQ[t]; int cnt = 0;
        for (int j = 0; j < NC; ++j) { float sj = sQ[j]; cnt += (sj > st) || (sj == st && j < t); }
        if (cnt == 31) thrQ = st;                     // k = 32 (gQ and gK share it)
        st = sV[t]; cnt = 0;
        for (int j = 0; j < NC; ++j) { float sj = sV[j]; cnt += (sj > st) || (sj == st && j < t); }
        if (cnt == 63) thrV = st;                     // k = 64
    }
    __syncthreads();

    // ---- phase 6: gates
    if (t < NC) {
        const float sq = sQ[t], sv = sV[t];
        egQ[t] = (sq >= thrQ) ? gate_val(sq, proj[4]) : 0.0f;
        egK[t] = (sq >= thrQ) ? gate_val(sq, proj[5]) : 0.0f;
        egV[t] = (sv >= thrV) ? gate_val(sv, proj[6]) : 0.0f;
    }
    __syncthreads();
    if (t < 3) {
        const float* e = (t == 0) ? egQ : (t == 1) ? egK : egV;
        float sm = 0.0f, mx = -INFINITY;
        for (int j = 0; j < NC; ++j) { sm += e[j]; mx = fmaxf(mx, e[j]); }
        scl[t] = tanhf(mx) / (sm + 1e-8f);
    }
    __syncthreads();

    // ---- phase 7: write gates + per-block pos_loss partials (fixed-order tree)
    float4 part; part.x = part.y = part.z = part.w = 0.0f;
    if (t < NC) {
        const float gq = egQ[t] * scl[0];
        const float gk = egK[t] * scl[1];
        const float gv = egV[t] * scl[2];
        const size_t base = (size_t)token * NC + t;
        out[base]           = gq;
        out[GSZ + base]     = gk;
        out[2 * GSZ + base] = gv;
        if (ciQ[t] >= 0) {
            const int n = ciQ[t];
            const float dx = proj[0] - npos[2 * n], dy = proj[1] - npos[2 * n + 1];
            part.x = gq * (dx * dx + dy * dy);
            part.y = 1.0f;
        }
        if (ciV[t] >= 0) {
            const int n = NQK + ciV[t];
            const float dx = proj[2] - npos[2 * n], dy = proj[3] - npos[2 * n + 1];
            part.z = egV[t] * scl[2] * (dx * dx + dy * dy);
            part.w = 1.0f;
        }
    }
    red[t] = part;
    __syncthreads();
    for (int off = 128; off > 0; off >>= 1) {
        if (t < off) {
            float4 a = red[t], b = red[t + off];
            a.x += b.x; a.y += b.y; a.z += b.z; a.w += b.w;
            red[t] = a;
        }
        __syncthreads();
    }
    if (t == 0) {
        ws[0 * TOK + token] = red[0].x;   // sum gQ*d_qk*m
        ws[1 * TOK + token] = red[0].y;   // sum m_qk
        ws[2 * TOK + token] = red[0].z;   // sum gV*d_v*m
        ws[3 * TOK + token] = red[0].w;   // sum m_v
    }
}

__global__ __launch_bounds__(256) void loss_reduce(const float* __restrict__ ws,
                                                   float* __restrict__ out)
{
    __shared__ float4 red[256];
    const int t = threadIdx.x;
    float4 a; a.x = a.y = a.z = a.w = 0.0f;
    for (int i = t; i < TOK; i += 256) {
        a.x += ws[i];
        a.y += ws[TOK + i];
        a.z += ws[2 * TOK + i];
        a.w += ws[3 * TOK + i];
    }
    red[t] = a;
    __syncthreads();
    for (int off = 128; off > 0; off >>= 1) {
        if (t < off) {
            float4 p = red[t], q = red[t + off];
            p.x += q.x; p.y += q.y; p.z += q.z; p.w += q.w;
            red[t] = p;
        }
        __syncthreads();
    }
    if (t == 0) {
        out[3 * GSZ] = red[0].x / (red[0].y + 1e-8f) + red[0].z / (red[0].w + 1e-8f);
    }
}

extern "C" void kernel_launch(void* const* d_in, const int* in_sizes, int n_in,
                              void* d_out, int out_size, void* d_ws, size_t ws_size,
                              hipStream_t stream) {
    const float* x    = (const float*)d_in[0];
    const float* qkN  = (const float*)d_in[1];
    const float* vN   = (const float*)d_in[2];
    const float* npos = (const float*)d_in[3];
    const float* Wpq  = (const float*)d_in[4];
    const float* bpq  = (const float*)d_in[5];
    const float* Wpv  = (const float*)d_in[6];
    const float* bpv  = (const float*)d_in[7];
    const float* Wtau = (const float*)d_in[8];
    const float* btau = (const float*)d_in[9];
    const int*   cmQ  = (const int*)d_in[10];
    const int*   cmV  = (const int*)d_in[11];
    float* out = (float*)d_out;
    float* ws  = (float*)d_ws;

    router_fwd<<<TOK, 256, 0, stream>>>(x, qkN, vN, npos, Wpq, bpq, Wpv, bpv,
                                        Wtau, btau, cmQ, cmV, out, ws);
    loss_reduce<<<1, 256, 0, stream>>>(ws, out);
}